// GatCodeNet_4398046511490
// MI455X (gfx1250) — compile-verified
//
#include <hip/hip_runtime.h>
#include <hip/hip_bf16.h>
#include <math.h>

#define N_NODES   100000
#define N_EDGES   400000
#define E_TOT     (N_EDGES + N_NODES)   // with self loops
#define HIDDEN    256
#define N_LAYERS  4
#define MAX_DEPTH 20
#define NUM_GRAPHS 128
#define SEQ_LEN   5
#define VOCAB     5002
#define NEG_SLOPE 0.2f
#define BN_EPS    1e-5f
#define PANEL_COLS 32                   // B panel width staged to LDS per block

typedef __attribute__((ext_vector_type(2))) float v2f;
typedef __attribute__((ext_vector_type(8))) float v8f;
typedef __attribute__((ext_vector_type(4))) unsigned int u32x4;
typedef __attribute__((ext_vector_type(4))) int i32x4;
typedef __attribute__((ext_vector_type(8))) int i32x8;

// ---------------------------------------------------------------- utilities
__global__ void init_f32_kernel(float* __restrict__ p, float v, int n) {
    int i = blockIdx.x * blockDim.x + threadIdx.x;
    if (i < n) p[i] = v;
}

__device__ inline void atomicMaxF(float* addr, float val) {
    unsigned* ua = (unsigned*)addr;
    unsigned old = __float_as_uint(*addr);
    while (__uint_as_float(old) < val) {
        unsigned assumed = old;
        old = atomicCAS(ua, assumed, __float_as_uint(val));
        if (old == assumed) break;
    }
}

__device__ inline void edge_ends(const int* __restrict__ ei, int e, int& s, int& d) {
    if (e < N_EDGES) { s = ei[e]; d = ei[N_EDGES + e]; }
    else             { s = e - N_EDGES; d = s; }
}

// ---------------------------------------------------------------- embedding
__global__ void embed_kernel(const int* __restrict__ x, const int* __restrict__ depth,
                             const float* __restrict__ type_emb,
                             const float* __restrict__ attr_emb,
                             const float* __restrict__ depth_emb,
                             float* __restrict__ h) {
    long long i = (long long)blockIdx.x * blockDim.x + threadIdx.x;
    if (i >= (long long)N_NODES * HIDDEN) return;
    int n = (int)(i >> 8), c = (int)(i & 255);
    int t = x[2 * n], a = x[2 * n + 1];
    int dp = depth[n]; dp = dp > MAX_DEPTH ? MAX_DEPTH : dp;
    h[i] = type_emb[t * HIDDEN + c] + attr_emb[a * HIDDEN + c] + depth_emb[dp * HIDDEN + c];
}

// -------------------------------------------------- fp32 WMMA tiled GEMM
// C[M,N] = A[M,K=HIDDEN] @ B[K,N] (+ bias[N]).
// Block = 8 waves covering 128 rows x one 32-col panel.  The B panel
// (K x 32, row stride N) is DMA'd into LDS once per block by the Tensor
// Data Mover (tensor_load_to_lds, TENSORcnt), with hardware OOB zero-fill
// covering the ragged N=5002 edge.  Each wave then runs the K-loop:
// A fragment via global_load_b64 (row-clamped; row m only feeds C row m),
// B fragments via ds_load from the staged panel, two V_WMMA_F32_16X16X4_F32
// per k-step (16x32 tile, shared A fragment).  Stores are guarded.
__global__ void gemm_wmma_f32_kernel(const float* __restrict__ A,
                                     const float* __restrict__ Bb,
                                     float* __restrict__ Cb,
                                     const float* __restrict__ biasb,
                                     int M, int N,
                                     long long bStride, long long cStride, long long biasStride) {
    const int K = HIDDEN;
    __shared__ float ldsB[HIDDEN * PANEL_COLS];      // 32 KB panel

    const float* B = Bb + (long long)blockIdx.y * bStride;
    float* C = Cb + (long long)blockIdx.y * cStride;
    const float* bias = biasb ? biasb + (long long)blockIdx.y * biasStride : nullptr;

    int panels = (N + PANEL_COLS - 1) / PANEL_COLS;
    int mblk = blockIdx.x / panels;
    int pn   = blockIdx.x % panels;
    int col0 = pn * PANEL_COLS;
    int wv   = threadIdx.x >> 5;
    int lane = threadIdx.x & 31;
    int row0 = mblk * 128 + wv * 16;

    // ---- TDM: async DMA of B panel [K x PANEL_COLS] into LDS (wave 0 only)
    if (threadIdx.x < 32) {
        unsigned lds_off = (unsigned)(size_t)ldsB;             // low 32 bits = LDS offset
        unsigned long long ga = (unsigned long long)(size_t)(B + col0);
        unsigned td0 = (unsigned)(N - col0);                   // valid cols; OOB reads -> 0

        u32x4 g0; i32x8 g1; i32x4 g2, g3;
        g0[0] = 1u;                                            // count=1, user descriptor
        g0[1] = lds_off;                                       // lds_addr
        g0[2] = (unsigned)ga;                                  // global_addr[31:0]
        g0[3] = (unsigned)((ga >> 32) & 0x1FFFFFFull) | (2u << 30);  // addr[56:32] | type=2

        g1[0] = (int)(2u << 16);                               // wg_mask=0, data_size=4B
        g1[1] = (int)((td0 & 0xFFFFu) << 16);                  // tensor_dim0[15:0]
        g1[2] = (int)(((td0 >> 16) & 0xFFFFu) |
                      (((unsigned)K & 0xFFFFu) << 16));        // dim0[31:16] | dim1[15:0]
        g1[3] = (int)((((unsigned)K >> 16) & 0xFFFFu) |
                      ((unsigned)PANEL_COLS << 16));           // dim1[31:16] | tile_dim0
        g1[4] = (int)((unsigned)K & 0xFFFFu);                  // tile_dim1=K, tile_dim2=0
        g1[5] = N;                                             // tensor_dim0_stride[31:0]
        g1[6] = 0;                                             // stride hi | dim1_stride lo
        g1[7] = 0;
        g2[0] = g2[1] = g2[2] = g2[3] = 0;                     // 2D tile: dims 2..4 unused
        g3[0] = g3[1] = g3[2] = g3[3] = 0;
#if defined(__clang_major__) && __clang_major__ >= 23
        i32x8 g4; g4[0]=g4[1]=g4[2]=g4[3]=g4[4]=g4[5]=g4[6]=g4[7]=0;
        __builtin_amdgcn_tensor_load_to_lds(g0, g1, g2, g3, g4, 0);
#else
        __builtin_amdgcn_tensor_load_to_lds(g0, g1, g2, g3, 0);
#endif
        __builtin_amdgcn_s_wait_tensorcnt(0);
    }
    __syncthreads();

    int l  = lane & 15;                    // A row / B col within tile
    int kh = (lane >> 4) << 1;             // K sub-offset: 0 (lanes 0-15) / 2 (16-31)
    int ar = row0 + l; if (ar > M - 1) ar = M - 1;   // clamp: row m only feeds C row m

    const float* Ap = A + (long long)ar * K + kh;    // 8B aligned
    const float* Lp = ldsB + kh * PANEL_COLS;

    v8f c0 = {}, c1 = {};
    #pragma unroll 4
    for (int k0 = 0; k0 < K; k0 += 4) {
        v2f a = *(const v2f*)(Ap + k0);              // global_load_b64
        const float* Lk = Lp + k0 * PANEL_COLS;
        v2f b0, b1;
        b0[0] = Lk[l];                b0[1] = Lk[PANEL_COLS + l];
        b1[0] = Lk[16 + l];           b1[1] = Lk[PANEL_COLS + 16 + l];
        if (k0 + 64 < K) __builtin_prefetch(Ap + k0 + 64, 0, 3);
        c0 = __builtin_amdgcn_wmma_f32_16x16x4_f32(false, a, false, b0,
                                                   (short)0, c0, false, false);
        c1 = __builtin_amdgcn_wmma_f32_16x16x4_f32(false, a, false, b1,
                                                   (short)0, c1, false, false);
    }

    // C layout: lane holds column; VGPR r holds row row0 + r + 8*(lane>>4)
    int mbase = row0 + ((lane >> 4) << 3);
    int nc0 = col0 + l, nc1 = col0 + 16 + l;
    int bb0 = nc0 < N ? nc0 : N - 1, bb1 = nc1 < N ? nc1 : N - 1;
    float bv0 = bias ? bias[bb0] : 0.f;
    float bv1 = bias ? bias[bb1] : 0.f;
    #pragma unroll
    for (int r = 0; r < 8; ++r) {
        int mr = mbase + r;
        if (mr < M) {
            long long rowoff = (long long)mr * N;
            if (nc0 < N) C[rowoff + nc0] = c0[r] + bv0;
            if (nc1 < N) C[rowoff + nc1] = c1[r] + bv1;
        }
    }
}

// ---------------------------------------------------------- attention scores
__global__ void att_score_kernel(const float* __restrict__ hh,
                                 const float* __restrict__ a_src,
                                 const float* __restrict__ a_dst,
                                 float* __restrict__ s_src, float* __restrict__ s_dst,
                                 int heads) {
    int i = blockIdx.x * blockDim.x + threadIdx.x;
    if (i >= N_NODES * heads) return;
    int n = i / heads, hd = i % heads;
    int d = HIDDEN / heads;
    const float* hp = hh + (long long)n * HIDDEN + hd * d;
    float ss = 0.f, sd = 0.f;
    for (int j = 0; j < d; ++j) {
        float v = hp[j];
        ss += v * a_src[hd * d + j];
        sd += v * a_dst[hd * d + j];
    }
    s_src[i] = ss; s_dst[i] = sd;
}

// ------------------------------------------------------------ edge passes
__global__ void edge_max_kernel(const int* __restrict__ ei,
                                const float* __restrict__ s_src,
                                const float* __restrict__ s_dst,
                                float* __restrict__ m, int heads) {
    long long i = (long long)blockIdx.x * blockDim.x + threadIdx.x;
    if (i >= (long long)E_TOT * heads) return;
    int e = (int)(i / heads), hd = (int)(i % heads);
    int s, d; edge_ends(ei, e, s, d);
    float v = s_src[s * heads + hd] + s_dst[d * heads + hd];
    v = v > 0.f ? v : NEG_SLOPE * v;
    atomicMaxF(&m[d * heads + hd], v);
}

__global__ void edge_expsum_kernel(const int* __restrict__ ei,
                                   const float* __restrict__ s_src,
                                   const float* __restrict__ s_dst,
                                   const float* __restrict__ m,
                                   float* __restrict__ denom, int heads) {
    long long i = (long long)blockIdx.x * blockDim.x + threadIdx.x;
    if (i >= (long long)E_TOT * heads) return;
    int e = (int)(i / heads), hd = (int)(i % heads);
    int s, d; edge_ends(ei, e, s, d);
    float v = s_src[s * heads + hd] + s_dst[d * heads + hd];
    v = v > 0.f ? v : NEG_SLOPE * v;
    atomicAdd(&denom[d * heads + hd], __expf(v - m[d * heads + hd]));
}

__global__ void edge_aggregate_kernel(const int* __restrict__ ei,
                                      const float* __restrict__ hh,
                                      const float* __restrict__ s_src,
                                      const float* __restrict__ s_dst,
                                      const float* __restrict__ m,
                                      const float* __restrict__ denom,
                                      float* __restrict__ agg, int heads) {
    long long i = (long long)blockIdx.x * blockDim.x + threadIdx.x;
    if (i >= (long long)E_TOT * HIDDEN) return;
    int e = (int)(i >> 8), c = (int)(i & 255);
    int dmul = HIDDEN / heads;
    int hd = c / dmul;
    int s, d; edge_ends(ei, e, s, d);
    float v = s_src[s * heads + hd] + s_dst[d * heads + hd];
    v = v > 0.f ? v : NEG_SLOPE * v;
    float alpha = __expf(v - m[d * heads + hd]) / denom[d * heads + hd];
    atomicAdd(&agg[(long long)d * HIDDEN + c], hh[(long long)s * HIDDEN + c] * alpha);
}

// ---------------------------------------------------------------- batchnorm
__global__ void bn_stats_kernel(const float* __restrict__ agg,
                                float* __restrict__ mean, float* __restrict__ rstd) {
    int c = blockIdx.x;
    float s = 0.f, s2 = 0.f;
    for (int n = threadIdx.x; n < N_NODES; n += blockDim.x) {
        float v = agg[(long long)n * HIDDEN + c];
        s += v; s2 += v * v;
    }
    __shared__ float sh[256], sh2[256];
    sh[threadIdx.x] = s; sh2[threadIdx.x] = s2;
    __syncthreads();
    for (int off = 128; off > 0; off >>= 1) {
        if (threadIdx.x < off) {
            sh[threadIdx.x]  += sh[threadIdx.x + off];
            sh2[threadIdx.x] += sh2[threadIdx.x + off];
        }
        __syncthreads();
    }
    if (threadIdx.x == 0) {
        float mu  = sh[0] / (float)N_NODES;
        float var = sh2[0] / (float)N_NODES - mu * mu;
        mean[c] = mu;
        rstd[c] = rsqrtf(var + BN_EPS);
    }
}

__global__ void bn_apply_kernel(const float* __restrict__ agg,
                                const float* __restrict__ identity,
                                const float* __restrict__ mean,
                                const float* __restrict__ rstd,
                                const float* __restrict__ gamma,
                                const float* __restrict__ beta,
                                float* __restrict__ h_next) {
    long long i = (long long)blockIdx.x * blockDim.x + threadIdx.x;
    if (i >= (long long)N_NODES * HIDDEN) return;
    int c = (int)(i & 255);
    float v = (agg[i] - mean[c]) * rstd[c] * gamma[c] + beta[c];
    v = v > 0.f ? v : 0.f;
    h_next[i] = v + identity[i];
}

// ---------------------------------------------------------------- pooling
__global__ void pool_count_kernel(const int* __restrict__ batch, float* __restrict__ cnt) {
    int n = blockIdx.x * blockDim.x + threadIdx.x;
    if (n < N_NODES) atomicAdd(&cnt[batch[n]], 1.f);
}

__global__ void pool_sum_kernel(const int* __restrict__ batch,
                                const float* __restrict__ h, float* __restrict__ g) {
    long long i = (long long)blockIdx.x * blockDim.x + threadIdx.x;
    if (i >= (long long)N_NODES * HIDDEN) return;
    int n = (int)(i >> 8), c = (int)(i & 255);
    atomicAdd(&g[batch[n] * HIDDEN + c], h[i]);
}

__global__ void pool_div_kernel(float* __restrict__ g, const float* __restrict__ cnt) {
    int i = blockIdx.x * blockDim.x + threadIdx.x;
    if (i >= NUM_GRAPHS * HIDDEN) return;
    float c = cnt[i >> 8];
    g[i] /= (c > 1.f ? c : 1.f);
}

// ================================================================ launcher
extern "C" void kernel_launch(void* const* d_in, const int* in_sizes, int n_in,
                              void* d_out, int out_size, void* d_ws, size_t ws_size,
                              hipStream_t stream) {
    const int*   x          = (const int*)  d_in[0];
    const int*   node_depth = (const int*)  d_in[1];
    const int*   edge_index = (const int*)  d_in[2];
    const int*   batch      = (const int*)  d_in[3];
    const float* type_emb   = (const float*)d_in[4];
    const float* attr_emb   = (const float*)d_in[5];
    const float* depth_emb  = (const float*)d_in[6];
    const float* Ws         = (const float*)d_in[7];
    const float* att_src    = (const float*)d_in[8];
    const float* att_dst    = (const float*)d_in[9];
    // d_in[10] gat_bias (zeros; cancels through batchnorm) -- unused
    const float* bn_gamma   = (const float*)d_in[11];
    const float* bn_beta    = (const float*)d_in[12];
    const float* tok_W      = (const float*)d_in[13];
    const float* tok_b      = (const float*)d_in[14];
    float* out = (float*)d_out;

    // ---- workspace carve-up
    char* w = (char*)d_ws;
    const size_t szH = (size_t)N_NODES * HIDDEN * sizeof(float);
    float* bufs[3] = { (float*)w, (float*)(w + szH), (float*)(w + 2 * szH) };
    w += 3 * szH;
    float* s_src = (float*)w; w += (size_t)N_NODES * 8 * sizeof(float);
    float* s_dst = (float*)w; w += (size_t)N_NODES * 8 * sizeof(float);
    float* mbuf  = (float*)w; w += (size_t)N_NODES * 8 * sizeof(float);
    float* dbuf  = (float*)w; w += (size_t)N_NODES * 8 * sizeof(float);
    float* meanb = (float*)w; w += HIDDEN * sizeof(float);
    float* rstdb = (float*)w; w += HIDDEN * sizeof(float);
    float* gbuf  = (float*)w; w += (size_t)NUM_GRAPHS * HIDDEN * sizeof(float);
    float* cntb  = (float*)w; w += NUM_GRAPHS * sizeof(float);

    const int TB = 256;
    const long long NH = (long long)N_NODES * HIDDEN;
    const int gNH = (int)((NH + TB - 1) / TB);

    // ---- embedding
    embed_kernel<<<gNH, TB, 0, stream>>>(x, node_depth, type_emb, attr_emb, depth_emb, bufs[0]);

    int cur = 0, t1 = 1, t2 = 2;
    for (int l = 0; l < N_LAYERS; ++l) {
        const int heads = (l == N_LAYERS - 1) ? 1 : 8;
        float* h_cur = bufs[cur];
        float* hh    = bufs[t1];
        float* agg   = bufs[t2];

        // hh = h_cur @ Ws[l]   (TDM-staged B panel + WMMA fp32)
        {
            int row_blocks = (N_NODES + 127) / 128;
            int panels = (HIDDEN + PANEL_COLS - 1) / PANEL_COLS;
            gemm_wmma_f32_kernel<<<dim3(row_blocks * panels, 1), TB, 0, stream>>>(
                h_cur, Ws + (long long)l * HIDDEN * HIDDEN, hh, nullptr,
                N_NODES, HIDDEN, 0, 0, 0);
        }

        // per-head scores
        {
            int n = N_NODES * heads;
            att_score_kernel<<<(n + TB - 1) / TB, TB, 0, stream>>>(
                hh, att_src + (long long)l * HIDDEN, att_dst + (long long)l * HIDDEN,
                s_src, s_dst, heads);
        }

        // init segment buffers
        {
            int nh = N_NODES * heads;
            init_f32_kernel<<<(nh + TB - 1) / TB, TB, 0, stream>>>(mbuf, -3.0e38f, nh);
            init_f32_kernel<<<(nh + TB - 1) / TB, TB, 0, stream>>>(dbuf, 0.f, nh);
            init_f32_kernel<<<gNH, TB, 0, stream>>>(agg, 0.f, (int)NH);
        }

        // softmax over incoming edges + weighted aggregation
        {
            long long eh = (long long)E_TOT * heads;
            int geh = (int)((eh + TB - 1) / TB);
            edge_max_kernel<<<geh, TB, 0, stream>>>(edge_index, s_src, s_dst, mbuf, heads);
            edge_expsum_kernel<<<geh, TB, 0, stream>>>(edge_index, s_src, s_dst, mbuf, dbuf, heads);
            long long ec = (long long)E_TOT * HIDDEN;
            int gec = (int)((ec + TB - 1) / TB);
            edge_aggregate_kernel<<<gec, TB, 0, stream>>>(edge_index, hh, s_src, s_dst,
                                                          mbuf, dbuf, agg, heads);
        }

        // batchnorm + relu + residual -> next h (reuses hh buffer)
        bn_stats_kernel<<<HIDDEN, 256, 0, stream>>>(agg, meanb, rstdb);
        bn_apply_kernel<<<gNH, TB, 0, stream>>>(agg, h_cur, meanb, rstdb,
                                                bn_gamma + (long long)l * HIDDEN,
                                                bn_beta + (long long)l * HIDDEN, hh);
        int nc = t1; t1 = t2; t2 = cur; cur = nc;
    }

    // ---- mean pool per graph
    init_f32_kernel<<<(NUM_GRAPHS * HIDDEN + TB - 1) / TB, TB, 0, stream>>>(gbuf, 0.f, NUM_GRAPHS * HIDDEN);
    init_f32_kernel<<<1, NUM_GRAPHS, 0, stream>>>(cntb, 0.f, NUM_GRAPHS);
    pool_count_kernel<<<(N_NODES + TB - 1) / TB, TB, 0, stream>>>(batch, cntb);
    pool_sum_kernel<<<gNH, TB, 0, stream>>>(batch, bufs[cur], gbuf);
    pool_div_kernel<<<(NUM_GRAPHS * HIDDEN + TB - 1) / TB, TB, 0, stream>>>(gbuf, cntb);

    // ---- token predictors: out[s] = g @ tok_W[s] + tok_b[s]   (TDM + WMMA fp32)
    {
        int row_blocks = (NUM_GRAPHS + 127) / 128;     // = 1
        int panels = (VOCAB + PANEL_COLS - 1) / PANEL_COLS;
        gemm_wmma_f32_kernel<<<dim3(row_blocks * panels, SEQ_LEN), TB, 0, stream>>>(
            gbuf, tok_W, out, tok_b,
            NUM_GRAPHS, VOCAB,
            (long long)HIDDEN * VOCAB, (long long)NUM_GRAPHS * VOCAB, (long long)VOCAB);
    }
}